// SCGATConvolution_56040733278450
// MI455X (gfx1250) — compile-verified
//
#include <hip/hip_runtime.h>

// SCGAT convolution for MI455X (gfx1250, wave32).
// Decomposition: emb_e = (x_src@W1)[s] + (x_tgt@W2)[t] + e_attr@W3 + b_att
// Node GEMMs use V_WMMA_F32_16X16X4_F32 with the weight slice staged in LDS
// (packed for single ds_load_b64 B-fragments). Edge stage is bandwidth/atomic
// bound and uses float4 gathers + LDS W3 + wave shuffles + global atomics.

typedef float v2f __attribute__((ext_vector_type(2)));
typedef float v8f __attribute__((ext_vector_type(8)));

#define HEADS   8
#define DIMV    32
#define C_TOT   256      // HEADS * DIMV
#define KF      128
#define F_EDGE  16
#define SLOPE   0.2f

__device__ __forceinline__ void atomicMaxF(float* addr, float v) {
  // bit-trick float atomic max (correct for mixed signs; init must be -inf)
  if (v >= 0.0f) atomicMax((int*)addr, __float_as_int(v));
  else           atomicMin((unsigned int*)addr, __float_as_uint(v));
}

// ---------------- K0: init out=bias, node_max=-inf, node_sum=0 ----------------
__global__ void k_init(float* __restrict__ out, const float* __restrict__ bias,
                       float* __restrict__ nmax, float* __restrict__ nsum, int N) {
  int i = blockIdx.x * blockDim.x + threadIdx.x;
  if (i < N * C_TOT) out[i] = bias[i & (C_TOT - 1)];
  if (i < N * HEADS) { nmax[i] = -__builtin_inff(); nsum[i] = 0.0f; }
}

// ---------------- K1: Y[M,Ncols] = X[M,128] @ W[128,Ncols] (+bias) ------------
// Block = 8 waves sharing one column group of S = 16*CT columns.
// W column-slice staged in LDS, packed as Wl[((k>>2)*S + col)*4 + (k&3)] so the
// (ka, ka+1) pair for lane half `half` is one contiguous ds_load_b64.
// A frag (16x4 f32): lane&15 = M-row, (lane>>4)*2 = K pair -> global_load_b64.
// C/D (16x16 f32): VGPR r, lane l -> row = r + 8*(l>>4), col = l&15.
// Wave-granular row guard AFTER the barrier keeps EXEC all-ones for WMMA.
template<int CT>
__global__ void k_gemm_wmma(const float* __restrict__ X, const float* __restrict__ W,
                            const float* __restrict__ bias, float* __restrict__ Y,
                            int M, int Ncols) {
  constexpr int S = 16 * CT;
  __shared__ float Wl[KF * S];

  int colGroups = Ncols / S;
  int blockRow  = blockIdx.x / colGroups;
  int cg        = blockIdx.x - blockRow * colGroups;

  // cooperative stage of the W column slice (coalesced global reads)
  for (int i = threadIdx.x; i < KF * S; i += blockDim.x) {
    int k = i / S, col = i - k * S;
    Wl[((k >> 2) * S + col) * 4 + (k & 3)] = W[(size_t)k * Ncols + cg * S + col];
  }
  __syncthreads();

  int w       = threadIdx.x >> 5;
  int lane    = threadIdx.x & 31;
  int rowTile = blockRow * 8 + w;
  if (rowTile >= (M >> 4)) return;   // whole wave exits -> EXEC stays all-ones

  int half = lane >> 4;              // 0 or 1
  int lmod = lane & 15;
  int m    = rowTile * 16 + lmod;

  v8f acc[CT];
  int ncol[CT];
#pragma unroll
  for (int t = 0; t < CT; ++t) {
    ncol[t] = cg * S + t * 16 + lmod;
    float bv = bias ? bias[ncol[t]] : 0.0f;
#pragma unroll
    for (int r = 0; r < 8; ++r) acc[t][r] = bv;
  }

  const float* __restrict__ xrow = X + (size_t)m * KF + half * 2;
  const v2f*   __restrict__ wl2  = (const v2f*)Wl;   // 8-byte pairs

  for (int k0 = 0; k0 < KF; k0 += 4) {
    v2f a = *(const v2f*)(xrow + k0);                // global_load_b64
    int kg = (k0 >> 2) * S;
#pragma unroll
    for (int t = 0; t < CT; ++t) {
      v2f b = wl2[(kg + t * 16 + lmod) * 2 + half];  // ds_load_b64
      acc[t] = __builtin_amdgcn_wmma_f32_16x16x4_f32(
          false, a, false, b, (short)0, acc[t], false, false);
    }
  }

#pragma unroll
  for (int t = 0; t < CT; ++t) {
#pragma unroll
    for (int r = 0; r < 8; ++r) {
      int row = rowTile * 16 + half * 8 + r;
      Y[(size_t)row * Ncols + ncol[t]] = acc[t][r];
    }
  }
}

// ---------------- K2: per-edge attention logits + atomic segment max ----------
// One wave per edge. Lane l owns channels c = l*8 .. l*8+7 (head = l>>2).
__global__ void k_edge_alpha(const float* __restrict__ Apro, const float* __restrict__ Bpro,
                             const long long* __restrict__ idx_src,
                             const long long* __restrict__ idx_tgt,
                             const float* __restrict__ e_attr,
                             const float* __restrict__ W3,   // [16][256], contiguous
                             const float* __restrict__ att,  // [256]
                             float* __restrict__ alpha, float* __restrict__ nmax, int E) {
  __shared__ float s_w3[F_EDGE * C_TOT];
  __shared__ float s_att[C_TOT];
  for (int i = threadIdx.x; i < F_EDGE * C_TOT; i += blockDim.x) s_w3[i] = W3[i];
  if (threadIdx.x < C_TOT) s_att[threadIdx.x] = att[threadIdx.x];
  __syncthreads();

  int e = blockIdx.x * (blockDim.x >> 5) + (threadIdx.x >> 5);
  if (e >= E) return;
  int lane = threadIdx.x & 31;
  long long s = idx_src[e], t = idx_tgt[e];
  int cbase = lane * 8;

  const float4* ap = (const float4*)(Apro + (size_t)s * C_TOT + cbase);
  const float4* bp = (const float4*)(Bpro + (size_t)t * C_TOT + cbase);
  float4 a0 = ap[0], a1 = ap[1], b0 = bp[0], b1 = bp[1];
  float emb[8] = {a0.x + b0.x, a0.y + b0.y, a0.z + b0.z, a0.w + b0.w,
                  a1.x + b1.x, a1.y + b1.y, a1.z + b1.z, a1.w + b1.w};

  float ea_l = (lane < F_EDGE) ? e_attr[(size_t)e * F_EDGE + lane] : 0.0f;
#pragma unroll
  for (int k = 0; k < F_EDGE; ++k) {
    float eak = __shfl(ea_l, k, 32);
    const float4* w4 = (const float4*)(s_w3 + k * C_TOT + cbase);
    float4 w0 = w4[0], w1 = w4[1];
    emb[0] = fmaf(eak, w0.x, emb[0]);
    emb[1] = fmaf(eak, w0.y, emb[1]);
    emb[2] = fmaf(eak, w0.z, emb[2]);
    emb[3] = fmaf(eak, w0.w, emb[3]);
    emb[4] = fmaf(eak, w1.x, emb[4]);
    emb[5] = fmaf(eak, w1.y, emb[5]);
    emb[6] = fmaf(eak, w1.z, emb[6]);
    emb[7] = fmaf(eak, w1.w, emb[7]);
  }

  float p = 0.0f;
#pragma unroll
  for (int j = 0; j < 8; ++j) {
    float v = emb[j];
    v = v > 0.0f ? v : SLOPE * v;          // leaky relu
    p = fmaf(v, s_att[cbase + j], p);
  }
  // reduce the 4 lanes of each head (lanes 4h..4h+3)
  p += __shfl_xor(p, 1, 32);
  p += __shfl_xor(p, 2, 32);

  if ((lane & 3) == 0) {
    int h = lane >> 2;
    alpha[(size_t)e * HEADS + h] = p;
    atomicMaxF(&nmax[(size_t)s * HEADS + h], p);
  }
}

// ---------------- K3: exp(alpha - max) and atomic segment sum -----------------
__global__ void k_softmax_norm(const long long* __restrict__ idx_src,
                               const float* __restrict__ nmax,
                               float* __restrict__ alpha, float* __restrict__ nsum, int E) {
  int i = blockIdx.x * blockDim.x + threadIdx.x;
  if (i >= E * HEADS) return;
  int e = i >> 3, h = i & 7;
  long long s = idx_src[e];
  float ev = __expf(alpha[i] - nmax[(size_t)s * HEADS + h]);
  alpha[i] = ev;
  atomicAdd(&nsum[(size_t)s * HEADS + h], ev);
}

// ---------------- K4: weighted scatter-add into out ---------------------------
// One wave per edge; lane = d (0..31). out already holds bias.
__global__ void k_scatter(const float* __restrict__ Vp,
                          const long long* __restrict__ idx_src,
                          const long long* __restrict__ idx_tgt,
                          const float* __restrict__ alpha, const float* __restrict__ nsum,
                          float* __restrict__ out, int E) {
  int e = blockIdx.x * (blockDim.x >> 5) + (threadIdx.x >> 5);
  if (e >= E) return;
  int lane = threadIdx.x & 31;
  long long s = idx_src[e], t = idx_tgt[e];
  float v = Vp[(size_t)t * DIMV + lane];
  float w = 0.0f;
  if (lane < HEADS) w = alpha[(size_t)e * HEADS + lane] / nsum[(size_t)s * HEADS + lane];
  float* op = out + (size_t)s * C_TOT + lane;
#pragma unroll
  for (int h = 0; h < HEADS; ++h) {
    float wh = __shfl(w, h, 32);
    atomicAdd(op + h * DIMV, wh * v);
  }
}

extern "C" void kernel_launch(void* const* d_in, const int* in_sizes, int n_in,
                              void* d_out, int out_size, void* d_ws, size_t ws_size,
                              hipStream_t stream) {
  const float*     x_src  = (const float*)d_in[0];
  const float*     x_tgt  = (const float*)d_in[1];
  const long long* mei    = (const long long*)d_in[2];   // int64 [2, E]
  const float*     e_attr = (const float*)d_in[3];
  const float*     W_att  = (const float*)d_in[4];       // [272, 256] row-major
  const float*     b_att  = (const float*)d_in[5];
  const float*     W_tgt  = (const float*)d_in[6];       // [128, 32]
  const float*     b_tgt  = (const float*)d_in[7];
  const float*     att    = (const float*)d_in[8];       // [1, 8, 32] -> 256
  const float*     bias   = (const float*)d_in[9];

  const int N = in_sizes[0] / KF;        // 50000 (multiple of 16)
  const int E = in_sizes[3] / F_EDGE;    // 800000
  const long long* idx_src = mei;
  const long long* idx_tgt = mei + E;

  // carve workspace (floats)
  float* ws = (float*)d_ws;
  size_t o = 0;
  float* Apro  = ws + o; o += (size_t)N * C_TOT;   // x_src @ W1 + b_att
  float* Bpro  = ws + o; o += (size_t)N * C_TOT;   // x_tgt @ W2
  float* Vp    = ws + o; o += (size_t)N * DIMV;    // x_tgt @ W_tgt + b_tgt
  float* alpha = ws + o; o += (size_t)E * HEADS;
  float* nmax  = ws + o; o += (size_t)N * HEADS;
  float* nsum  = ws + o; o += (size_t)N * HEADS;

  float* out = (float*)d_out;

  // K0: init
  {
    int threads = N * C_TOT;
    k_init<<<(threads + 255) / 256, 256, 0, stream>>>(out, bias, nmax, nsum, N);
  }

  // K1: node projections via WMMA (LDS-staged weight slices)
  {
    int rowTiles  = N / 16;                 // 3125
    int rowGroups = (rowTiles + 7) / 8;     // 8 row tiles (waves) per block
    int cg256     = C_TOT / 64;             // CT=4 -> 64 cols per block
    k_gemm_wmma<4><<<rowGroups * cg256, 256, 0, stream>>>(x_src, W_att,               b_att,   Apro, N, C_TOT);
    k_gemm_wmma<4><<<rowGroups * cg256, 256, 0, stream>>>(x_tgt, W_att + 128 * C_TOT, nullptr, Bpro, N, C_TOT);
    k_gemm_wmma<2><<<rowGroups * 1,     256, 0, stream>>>(x_tgt, W_tgt,               b_tgt,   Vp,   N, DIMV);
  }

  // K2..K4: edge pipeline
  int eblocks = (E + 7) / 8;   // 8 waves per block, one edge per wave
  k_edge_alpha<<<eblocks, 256, 0, stream>>>(Apro, Bpro, idx_src, idx_tgt, e_attr,
                                            W_att + 256 * C_TOT, att, alpha, nmax, E);
  k_softmax_norm<<<(E * HEADS + 255) / 256, 256, 0, stream>>>(idx_src, nmax, alpha, nsum, E);
  k_scatter<<<eblocks, 256, 0, stream>>>(Vp, idx_src, idx_tgt, alpha, nsum, out, E);
}